// GATEncoder_6004364280564
// MI455X (gfx1250) — compile-verified
//
#include <hip/hip_runtime.h>
#include <hip/hip_bf16.h>
#include <limits.h>

// ---------------------------------------------------------------------------
// GAT encoder for MI455X (gfx1250, wave32).
//  - Dense per-layer GEMM (h @ W_lin) on the f32 matrix pipe: V_WMMA_F32_16X16X4_F32
//  - Edge phase: per-edge atomics; xl/out (51.2 MB each) are L2-resident on the
//    192 MB L2, so random gather/scatter never touches HBM.
//  - a_edge folded to a 2x8 matrix (Me = W_edge . att_edge), alpha recomputed
//    per softmax pass (cheaper than materializing a 54 MB alpha array).
// ---------------------------------------------------------------------------

#define HID   128
#define HEADS 8
#define CPH   16
#define NLAYERS 3
#define NEG_SLOPE 0.2f
#define LN_EPS 1e-5f

typedef __attribute__((ext_vector_type(2))) float v2f;
typedef __attribute__((ext_vector_type(8))) float v8f;

__device__ __forceinline__ void atomAddF(float* p, float v) {
  __hip_atomic_fetch_add(p, v, __ATOMIC_RELAXED, __HIP_MEMORY_SCOPE_AGENT);
}
__device__ __forceinline__ void atomMaxI(int* p, int v) {
  __hip_atomic_fetch_max(p, v, __ATOMIC_RELAXED, __HIP_MEMORY_SCOPE_AGENT);
}
// monotone float<->int order-preserving map (involution)
__device__ __forceinline__ int fenc(float f) {
  int i = __float_as_int(f);
  return i >= 0 ? i : (i ^ 0x7FFFFFFF);
}
__device__ __forceinline__ float fdec(int i) {
  i = i >= 0 ? i : (i ^ 0x7FFFFFFF);
  return __int_as_float(i);
}
__device__ __forceinline__ float leaky(float x) { return x >= 0.f ? x : NEG_SLOPE * x; }
__device__ __forceinline__ float eluf(float x)  { return x > 0.f ? x : expm1f(x); }

// --------------------------- self-loop attr (mean) -------------------------
__global__ void k_zero(float* cnt, float* la, int N) {
  int n = blockIdx.x * blockDim.x + threadIdx.x;
  if (n < N) { cnt[n] = 0.f; la[2*n] = 0.f; la[2*n+1] = 0.f; }
}
__global__ void k_loop_accum(const int* __restrict__ dst, const float* __restrict__ ea,
                             float* cnt, float* la, int E) {
  int e = blockIdx.x * blockDim.x + threadIdx.x;
  if (e >= E) return;
  int d = dst[e];
  atomAddF(&cnt[d], 1.f);
  atomAddF(&la[2*d + 0], ea[2*e + 0]);
  atomAddF(&la[2*d + 1], ea[2*e + 1]);
}
__global__ void k_loop_final(float* la, const float* cnt, int N) {
  int n = blockIdx.x * blockDim.x + threadIdx.x;
  if (n >= N) return;
  float inv = 1.f / fmaxf(cnt[n], 1.f);
  la[2*n] *= inv; la[2*n+1] *= inv;
}

// ------------------------------- input proj --------------------------------
__global__ void k_input_proj(const float* __restrict__ x, const float* __restrict__ W,
                             const float* __restrict__ b, float* __restrict__ h, int N) {
  long t = (long)blockIdx.x * blockDim.x + threadIdx.x;
  if (t >= (long)N * HID) return;
  int n = (int)(t >> 7), c = (int)(t & 127);
  h[t] = b[c] + x[3*n]*W[c] + x[3*n+1]*W[HID + c] + x[3*n+2]*W[2*HID + c];
}

// --------------------- WMMA GEMM: xl = h @ W  (N x 128 @ 128 x 128) --------
// Block = 128 threads = 4 waves. Block computes a 16x128 tile; wave w does
// cols [32w, 32w+32) as two 16x16 accumulators. K loop in steps of 4 using
// V_WMMA_F32_16X16X4_F32 (full f32 precision).
// A-fragment layout (16x4 f32): v0 = K{0,2}, v1 = K{1,3}; lanes 0-15 M, 16-31 M again.
// D layout: VGPR v -> M=v (lanes 0-15) / M=v+8 (lanes 16-31), N = lane%16.
__global__ void __launch_bounds__(128) k_gemm_wmma(const float* __restrict__ A,
                                                   const float* __restrict__ W,
                                                   float* __restrict__ C, int N) {
  __shared__ float As[16][132];   // stride 132: conflict-free fragment reads
  const int tid = threadIdx.x;
  const int rowBase = blockIdx.x * 16;
  for (int i = tid; i < 16 * HID; i += 128) {
    int r = i >> 7, c = i & 127;
    int row = rowBase + r;
    As[r][c] = (row < N) ? A[(long)row * HID + c] : 0.f;
  }
  __syncthreads();

  const int lane   = tid & 31;
  const int laneLo = lane & 15;
  const int laneHi = lane >> 4;         // 0 or 1
  const int colBase = (tid >> 5) * 32;  // wave id * 32

  v8f acc0 = {};
  v8f acc1 = {};
  for (int k = 0; k < HID; k += 4) {
    v2f a, b0, b1;
    a.x = As[laneLo][k + 2*laneHi];
    a.y = As[laneLo][k + 2*laneHi + 1];
    const float* Wk = W + (long)(k + 2*laneHi) * HID + laneLo;
    b0.x = Wk[colBase];        b0.y = Wk[HID + colBase];        // rows K, K+1
    b1.x = Wk[colBase + 16];   b1.y = Wk[HID + colBase + 16];
    acc0 = __builtin_amdgcn_wmma_f32_16x16x4_f32(false, a, false, b0, (short)0, acc0, false, false);
    acc1 = __builtin_amdgcn_wmma_f32_16x16x4_f32(false, a, false, b1, (short)0, acc1, false, false);
  }

  const int rb = rowBase + 8 * laneHi;
  #pragma unroll
  for (int v = 0; v < 8; ++v) {
    int r = rb + v;
    if (r < N) {
      C[(long)r * HID + colBase + laneLo]      = acc0[v];
      C[(long)r * HID + colBase + 16 + laneLo] = acc1[v];
    }
  }
}

// ------------------- per-node per-head attention pre-scores ----------------
__global__ void k_att(const float* __restrict__ xl, const float* __restrict__ attS,
                      const float* __restrict__ attD, float* aS, float* aD, int N) {
  long t = (long)blockIdx.x * blockDim.x + threadIdx.x;
  if (t >= (long)N * HEADS) return;
  int n = (int)(t >> 3), hh = (int)(t & 7);
  const float* xp = xl + (long)n * HID + hh * CPH;
  const float* sp = attS + hh * CPH;
  const float* dp = attD + hh * CPH;
  float s = 0.f, d = 0.f;
  #pragma unroll
  for (int c = 0; c < CPH; ++c) { float v = xp[c]; s += v * sp[c]; d += v * dp[c]; }
  aS[t] = s; aD[t] = d;
}

// Me[d][h] = sum_c W_edge[d][h*16+c] * att_edge[h][c]  (folds a_edge to ea @ Me)
__global__ void k_me(const float* __restrict__ We, const float* __restrict__ ae, float* Me) {
  int t = threadIdx.x;
  if (t >= 2 * HEADS) return;
  int d = t >> 3, hh = t & 7;
  float s = 0.f;
  #pragma unroll
  for (int c = 0; c < CPH; ++c) s += We[d * HID + hh * CPH + c] * ae[hh * CPH + c];
  Me[t] = s;
}

__global__ void k_init_layer(float* outb, int* mEnc, float* ssum,
                             const float* __restrict__ bias, int N) {
  long t = (long)blockIdx.x * blockDim.x + threadIdx.x;
  if (t >= (long)N * HID) return;
  int c = (int)(t & 127);
  outb[t] = bias[c];
  if (c < HEADS) {
    int n = (int)(t >> 7);
    mEnc[n * HEADS + c] = INT_MIN;   // every node has a self-loop -> always updated
    ssum[n * HEADS + c] = 0.f;
  }
}

struct EdgeInfo { int s, d; float e0, e1; };
__device__ __forceinline__ EdgeInfo fetch_edge(int e, const int* src, const int* dst,
                                               const float* ea, const float* la, int E) {
  EdgeInfo r;
  if (e < E) { r.s = src[e]; r.d = dst[e]; r.e0 = ea[2*e]; r.e1 = ea[2*e+1]; }
  else { int n = e - E; r.s = n; r.d = n; r.e0 = la[2*n]; r.e1 = la[2*n+1]; }
  return r;
}
__device__ __forceinline__ float edge_alpha(const EdgeInfo& ei, int hh,
                                            const float* aS, const float* aD, const float* Me) {
  return leaky(aS[ei.s * HEADS + hh] + aD[ei.d * HEADS + hh]
               + ei.e0 * Me[hh] + ei.e1 * Me[HEADS + hh]);
}

// pass 1: scatter-max of alpha per (dst, head)
__global__ void k_pass1(const int* src, const int* dst, const float* ea, const float* la,
                        const float* __restrict__ aS, const float* __restrict__ aD,
                        const float* __restrict__ Me, int* mEnc, int E, int N) {
  int e = blockIdx.x * blockDim.x + threadIdx.x;
  if (e >= E + N) return;
  EdgeInfo ei = fetch_edge(e, src, dst, ea, la, E);
  #pragma unroll
  for (int hh = 0; hh < HEADS; ++hh)
    atomMaxI(&mEnc[ei.d * HEADS + hh], fenc(edge_alpha(ei, hh, aS, aD, Me)));
}

// pass 2: scatter-sum of exp(alpha - max)
__global__ void k_pass2(const int* src, const int* dst, const float* ea, const float* la,
                        const float* __restrict__ aS, const float* __restrict__ aD,
                        const float* __restrict__ Me, const int* __restrict__ mEnc,
                        float* ssum, int E, int N) {
  int e = blockIdx.x * blockDim.x + threadIdx.x;
  if (e >= E + N) return;
  EdgeInfo ei = fetch_edge(e, src, dst, ea, la, E);
  #pragma unroll
  for (int hh = 0; hh < HEADS; ++hh) {
    float al = edge_alpha(ei, hh, aS, aD, Me);
    float ex = __expf(al - fdec(mEnc[ei.d * HEADS + hh]));
    atomAddF(&ssum[ei.d * HEADS + hh], ex);
  }
}

// pass 3: one thread per (edge, head): gather 16 ch of xl[src], scale, scatter-add
__global__ void k_pass3(const int* src, const int* dst, const float* ea, const float* la,
                        const float* __restrict__ aS, const float* __restrict__ aD,
                        const float* __restrict__ Me, const int* __restrict__ mEnc,
                        const float* __restrict__ ssum, const float* __restrict__ xl,
                        float* outb, int E, int N) {
  long t = (long)blockIdx.x * blockDim.x + threadIdx.x;
  int e = (int)(t >> 3);
  if (e >= E + N) return;
  int hh = (int)(t & 7);
  EdgeInfo ei = fetch_edge(e, src, dst, ea, la, E);
  int di = ei.d * HEADS + hh;
  float al = edge_alpha(ei, hh, aS, aD, Me);
  float a = __expf(al - fdec(mEnc[di])) / (ssum[di] + 1e-16f);
  const float* xp = xl + (long)ei.s * HID + hh * CPH;
  float* op = outb + (long)ei.d * HID + hh * CPH;
  #pragma unroll
  for (int c = 0; c < CPH; ++c) atomAddF(&op[c], xp[c] * a);
}

// --------------------- LayerNorm -> ELU -> residual (wave32/node) ----------
__global__ void __launch_bounds__(256) k_ln_elu_res(const float* __restrict__ outb,
                                                    const float* __restrict__ hres,
                                                    const float* __restrict__ gamma,
                                                    const float* __restrict__ beta,
                                                    float* __restrict__ dstp, int N) {
  long gt = (long)blockIdx.x * blockDim.x + threadIdx.x;
  int node = (int)(gt >> 5);
  if (node >= N) return;
  int lane = (int)(gt & 31);
  float4 v = ((const float4*)(outb + (long)node * HID))[lane];
  float s = v.x + v.y + v.z + v.w;
  float q = v.x*v.x + v.y*v.y + v.z*v.z + v.w*v.w;
  #pragma unroll
  for (int off = 16; off >= 1; off >>= 1) {
    s += __shfl_xor(s, off, 32);
    q += __shfl_xor(q, off, 32);
  }
  float mu  = s * (1.f / HID);
  float var = q * (1.f / HID) - mu * mu;
  float inv = rsqrtf(var + LN_EPS);
  float4 g  = ((const float4*)gamma)[lane];
  float4 bb = ((const float4*)beta)[lane];
  float4 r  = ((const float4*)(hres + (long)node * HID))[lane];
  float4 o;
  o.x = eluf((v.x - mu) * inv * g.x + bb.x) + r.x;
  o.y = eluf((v.y - mu) * inv * g.y + bb.y) + r.y;
  o.z = eluf((v.z - mu) * inv * g.z + bb.z) + r.z;
  o.w = eluf((v.w - mu) * inv * g.w + bb.w) + r.w;
  ((float4*)(dstp + (long)node * HID))[lane] = o;
}

// ---------------------------------------------------------------------------
extern "C" void kernel_launch(void* const* d_in, const int* in_sizes, int n_in,
                              void* d_out, int out_size, void* d_ws, size_t ws_size,
                              hipStream_t stream) {
  (void)n_in; (void)out_size;
  const float* x        = (const float*)d_in[0];
  const int*   eidx     = (const int*)  d_in[1];
  const float* ea       = (const float*)d_in[2];
  const float* W_in     = (const float*)d_in[3];
  const float* b_in     = (const float*)d_in[4];
  const float* W_lin    = (const float*)d_in[5];
  const float* W_edge   = (const float*)d_in[6];
  const float* att_src  = (const float*)d_in[7];
  const float* att_dst  = (const float*)d_in[8];
  const float* att_edge = (const float*)d_in[9];
  const float* bias     = (const float*)d_in[10];
  const float* gamma    = (const float*)d_in[11];
  const float* beta     = (const float*)d_in[12];

  const int N = in_sizes[0] / 3;
  const int E = in_sizes[1] / 2;
  const int* src = eidx;
  const int* dst = eidx + E;

  // workspace carve-up (floats)
  float* wsf = (float*)d_ws;
  size_t off = 0;
  float* h    = wsf + off; off += (size_t)N * HID;
  float* xl   = wsf + off; off += (size_t)N * HID;
  float* outb = wsf + off; off += (size_t)N * HID;
  float* aS   = wsf + off; off += (size_t)N * HEADS;
  float* aD   = wsf + off; off += (size_t)N * HEADS;
  int*   mEnc = (int*)(wsf + off); off += (size_t)N * HEADS;
  float* ssum = wsf + off; off += (size_t)N * HEADS;
  float* la   = wsf + off; off += (size_t)2 * N;
  float* cnt  = wsf + off; off += (size_t)N;
  float* Me   = wsf + off; off += 16;
  if (ws_size < off * sizeof(float)) return;   // insufficient scratch: bail safely

  const int TB = 256;
  auto cdiv = [](long a, long b) { return (int)((a + b - 1) / b); };
  const long F = (long)E + N;  // fused edge count (edges + self loops)

  k_zero      <<<cdiv(N, TB), TB, 0, stream>>>(cnt, la, N);
  k_loop_accum<<<cdiv(E, TB), TB, 0, stream>>>(dst, ea, cnt, la, E);
  k_loop_final<<<cdiv(N, TB), TB, 0, stream>>>(la, cnt, N);
  k_input_proj<<<cdiv((long)N * HID, TB), TB, 0, stream>>>(x, W_in, b_in, h, N);

  for (int l = 0; l < NLAYERS; ++l) {
    const float* Wl   = W_lin    + (size_t)l * HID * HID;
    const float* We   = W_edge   + (size_t)l * 2 * HID;
    const float* asl  = att_src  + (size_t)l * HID;
    const float* adl  = att_dst  + (size_t)l * HID;
    const float* ael  = att_edge + (size_t)l * HID;
    const float* bl   = bias     + (size_t)l * HID;
    const float* gl   = gamma    + (size_t)l * HID;
    const float* btl  = beta     + (size_t)l * HID;

    k_me        <<<1, 32, 0, stream>>>(We, ael, Me);
    k_gemm_wmma <<<cdiv(N, 16), 128, 0, stream>>>(h, Wl, xl, N);
    k_att       <<<cdiv((long)N * HEADS, TB), TB, 0, stream>>>(xl, asl, adl, aS, aD, N);
    k_init_layer<<<cdiv((long)N * HID, TB), TB, 0, stream>>>(outb, mEnc, ssum, bl, N);
    k_pass1     <<<cdiv(F, TB), TB, 0, stream>>>(src, dst, ea, la, aS, aD, Me, mEnc, E, N);
    k_pass2     <<<cdiv(F, TB), TB, 0, stream>>>(src, dst, ea, la, aS, aD, Me, mEnc, ssum, E, N);
    k_pass3     <<<cdiv(F * HEADS, TB), TB, 0, stream>>>(src, dst, ea, la, aS, aD, Me, mEnc, ssum,
                                                         xl, outb, E, N);
    float* dstp = (l == NLAYERS - 1) ? (float*)d_out : h;
    k_ln_elu_res<<<cdiv((long)N * 32, TB), TB, 0, stream>>>(outb, h, gl, btl, dstp, N);
  }
}